// GAT_3461743640614
// MI455X (gfx1250) — compile-verified
//
#include <hip/hip_runtime.h>
#include <hip/hip_bf16.h>

// ---------------------------------------------------------------------------
// GAT link prediction on gfx1250 (MI455X).
// Pipeline: CSR build -> [WMMA GEMM -> attn dots -> edge softmax -> CSR
// aggregate] x2 -> decode dot products. fp32 end-to-end; GEMMs use
// V_WMMA_F32_16X16X4_F32 (full precision matrix pipe).
//
// Roofline: all working sets (h1 25.6MB, h2 51.2MB, edges ~7MB) fit in the
// 192MB L2, so the edge gather/aggregate phase (~1.3GB of L2 reads) dominates,
// not the 5 GFLOP of GEMM. fp32 WMMA keeps exactness; lower precision would
// buy nothing on a scatter/L2-bound problem.
// ---------------------------------------------------------------------------

#define NN 50000            // nodes (divisible by 16 -> 3125 row tiles)
#define EE 800000           // raw edges
#define ET (EE + NN)        // edges incl. self loops
#define EDN 100000          // decode edges (pos and neg each)

typedef float v2f __attribute__((ext_vector_type(2)));
typedef float v8f __attribute__((ext_vector_type(8)));

// ---------------- WMMA GEMM: C[M,NC] = A[M,K] @ B[K,NC], fp32 --------------
// One wave per 16x(16*NT) tile. grid.x = M/16, grid.y = NC/(16*NT),
// blockDim = 32. K and NC are compile-time so all loads use immediate
// offsets (no per-iteration 64-bit address math).
template <int K, int NC, int NT>
__global__ __launch_bounds__(32)
void gemm_wmma_f32(const float* __restrict__ A, const float* __restrict__ B,
                   float* __restrict__ C) {
  const int tm   = blockIdx.x << 4;
  const int tn   = blockIdx.y * (16 * NT);
  const int lane = threadIdx.x;
  const int half = lane >> 4;   // 0: K pair {0,1} / M rows 0-7 ; 1: {2,3} / 8-15
  const int idx  = lane & 15;

  // A 16x4 fragment (ISA 7.12.2): row = idx, ks = {2*half, 2*half+1}.
  const float* Ap = A + (size_t)(tm + idx) * K + 2 * half;   // 8B aligned
  // B 4x16 fragment (mirrored): col = idx, rows k+2*half, k+2*half+1.
  const float* Bp = B + (size_t)(2 * half) * NC + tn + idx;

  v8f acc[NT];
#pragma unroll
  for (int t = 0; t < NT; ++t) acc[t] = (v8f){};

#pragma unroll
  for (int k = 0; k < K; k += 4) {
    v2f a = *(const v2f*)(Ap + k);                 // one global_load_b64
#pragma unroll
    for (int t = 0; t < NT; ++t) {
      v2f b;
      b.x = Bp[(size_t)(k + 0) * NC + t * 16];     // immediate-offset b32
      b.y = Bp[(size_t)(k + 1) * NC + t * 16];
      acc[t] = __builtin_amdgcn_wmma_f32_16x16x4_f32(false, a, false, b,
                                                     (short)0, acc[t],
                                                     false, false);
    }
  }

#pragma unroll
  for (int t = 0; t < NT; ++t)
#pragma unroll
    for (int v = 0; v < 8; ++v)                    // D: row = vgpr + 8*half
      C[(size_t)(tm + (half << 3) + v) * NC + tn + t * 16 + idx] = acc[t][v];
}

// ------------- per-node attention logits: al = sum_c h[n,hd,c]*a[hd,c] -----
__global__ void attn_logits(const float* __restrict__ h,
                            const float* __restrict__ a_src,
                            const float* __restrict__ a_dst,
                            float* __restrict__ al_s, float* __restrict__ al_d,
                            int n_nodes, int H) {
  int t = blockIdx.x * blockDim.x + threadIdx.x;
  if (t >= n_nodes * H) return;
  int n = t / H, hd = t % H;
  const float* hp = h + (size_t)n * H * 32 + hd * 32;
  const float* as = a_src + hd * 32;
  const float* ad = a_dst + hd * 32;
  float s = 0.f, d = 0.f;
#pragma unroll 8
  for (int c = 0; c < 32; ++c) { float v = hp[c]; s += v * as[c]; d += v * ad[c]; }
  al_s[t] = s;
  al_d[t] = d;
}

// ---------------- CSR build: count -> scan -> scatter ----------------------
__global__ void count_dst(const int* __restrict__ ei_dst, int* __restrict__ cnt) {
  int e = blockIdx.x * blockDim.x + threadIdx.x;
  if (e >= ET) return;
  int dst = (e < EE) ? ei_dst[e] : (e - EE);
  atomicAdd(&cnt[dst], 1);
}

__global__ __launch_bounds__(1024)
void scan_counts(const int* __restrict__ cnt, int* __restrict__ row_start) {
  __shared__ int part[1024];
  const int tid   = threadIdx.x;
  const int chunk = (NN + 1023) / 1024;
  const int lo    = tid * chunk;
  const int hi    = (lo + chunk < NN) ? lo + chunk : NN;
  int s = 0;
  for (int i = lo; i < hi; ++i) s += cnt[i];
  part[tid] = s;
  __syncthreads();
  for (int off = 1; off < 1024; off <<= 1) {
    int v = 0;
    if (tid >= off) v = part[tid - off];
    __syncthreads();
    if (tid >= off) part[tid] += v;
    __syncthreads();
  }
  int base = (tid > 0) ? part[tid - 1] : 0;
  for (int i = lo; i < hi; ++i) { row_start[i] = base; base += cnt[i]; }
  if (tid == 0) row_start[NN] = part[1023];
}

__global__ void fill_eid(const int* __restrict__ ei_dst,
                         const int* __restrict__ row_start,
                         int* __restrict__ cursor, int* __restrict__ eid) {
  int e = blockIdx.x * blockDim.x + threadIdx.x;
  if (e >= ET) return;
  int dst = (e < EE) ? ei_dst[e] : (e - EE);
  int pos = atomicAdd(&cursor[dst], 1);
  eid[row_start[dst] + pos] = e;
}

// ---------------- edge softmax numerators + denominators -------------------
// (segment-max cancels analytically; logits are O(1) so plain exp is safe)
template <int H>
__global__ void edge_softmax_num(const int* __restrict__ ei_src,
                                 const int* __restrict__ ei_dst,
                                 const float* __restrict__ al_s,
                                 const float* __restrict__ al_d,
                                 float* __restrict__ ex,
                                 float* __restrict__ denom) {
  int t = blockIdx.x * blockDim.x + threadIdx.x;
  if (t >= ET * H) return;
  int e  = t / H;
  int hd = t % H;
  int src = (e < EE) ? ei_src[e] : (e - EE);
  int dst = (e < EE) ? ei_dst[e] : (e - EE);
  float v = al_s[src * H + hd] + al_d[dst * H + hd];
  v = (v > 0.f) ? v : 0.2f * v;           // LeakyReLU(0.2)
  float xv = __expf(v);
  ex[t] = xv;
  atomicAdd(&denom[dst * H + hd], xv);
}

// ---------------- gather-side aggregation over CSR -------------------------
// grid = NN, blockDim = H*32; thread owns one output channel.
template <int H, bool RELU>
__global__ void aggregate(const int* __restrict__ ei_src,
                          const int* __restrict__ row_start,
                          const int* __restrict__ eid,
                          const float* __restrict__ h,
                          const float* __restrict__ ex,
                          const float* __restrict__ denom,
                          const float* __restrict__ bias,
                          float* __restrict__ out) {
  const int n  = blockIdx.x;
  const int c  = threadIdx.x;
  const int hd = c >> 5;
  const int beg = row_start[n], end = row_start[n + 1];
  const float inv = 1.0f / (denom[n * H + hd] + 1e-16f);
  float acc = 0.f;
  for (int i = beg; i < end; ++i) {
    int e   = eid[i];
    int src = (e < EE) ? ei_src[e] : (e - EE);
    float alpha = ex[(size_t)e * H + hd] * inv;
    acc += h[(size_t)src * (H * 32) + c] * alpha;
  }
  acc += bias[c];
  if (RELU) acc = (acc > 0.f) ? acc : 0.f;
  out[(size_t)n * (H * 32) + c] = acc;
}

// ---------------- decode: logits = dot(z[i], z[j]) over 256 dims -----------
__global__ void decode_dot(const float* __restrict__ z,
                           const int* __restrict__ pos,
                           const int* __restrict__ neg,
                           float* __restrict__ out) {
  int wid  = (blockIdx.x * blockDim.x + threadIdx.x) >> 5;
  int lane = threadIdx.x & 31;
  if (wid >= 2 * EDN) return;
  int i, j;
  if (wid < EDN) { i = pos[wid];        j = pos[EDN + wid]; }
  else           { i = neg[wid - EDN];  j = neg[EDN + wid - EDN]; }
  const float* zi = z + (size_t)i * 256;
  const float* zj = z + (size_t)j * 256;
  float s = 0.f;
#pragma unroll
  for (int c = lane; c < 256; c += 32) s += zi[c] * zj[c];
#pragma unroll
  for (int off = 16; off > 0; off >>= 1) s += __shfl_down(s, off, 32);
  if (lane == 0) out[wid] = s;
}

// ---------------------------------------------------------------------------
static inline size_t al256(size_t x) { return (x + 255) & ~(size_t)255; }

extern "C" void kernel_launch(void* const* d_in, const int* in_sizes, int n_in,
                              void* d_out, int out_size, void* d_ws, size_t ws_size,
                              hipStream_t stream) {
  (void)in_sizes; (void)n_in; (void)out_size; (void)ws_size;

  const float* x   = (const float*)d_in[0];
  const int*   ei  = (const int*)  d_in[1];
  const int*   pe  = (const int*)  d_in[2];
  const int*   ne  = (const int*)  d_in[3];
  const float* W1  = (const float*)d_in[4];
  const float* as1 = (const float*)d_in[5];
  const float* ad1 = (const float*)d_in[6];
  const float* b1  = (const float*)d_in[7];
  const float* W2  = (const float*)d_in[8];
  const float* as2 = (const float*)d_in[9];
  const float* ad2 = (const float*)d_in[10];
  const float* b2  = (const float*)d_in[11];
  const int* e_src = ei;            // edge_index row 0
  const int* e_dst = ei + EE;       // edge_index row 1
  float* out = (float*)d_out;

  // ---- workspace carve ----
  char* w = (char*)d_ws;
  size_t off = 0;
  auto carve = [&](size_t bytes) { void* p = w + off; off += al256(bytes); return p; };
  float* h1    = (float*)carve((size_t)NN * 128 * 4);
  float* z1    = (float*)carve((size_t)NN * 128 * 4);
  float* h2    = (float*)carve((size_t)NN * 256 * 4);
  float* z2    = (float*)carve((size_t)NN * 256 * 4);
  float* al_s  = (float*)carve((size_t)NN * 8 * 4);
  float* al_d  = (float*)carve((size_t)NN * 8 * 4);
  float* denom = (float*)carve((size_t)NN * 8 * 4);
  float* ex    = (float*)carve((size_t)ET * 8 * 4);
  int*   rowst = (int*)  carve((size_t)(NN + 1) * 4);
  int*   cnt   = (int*)  carve((size_t)NN * 4);
  int*   eid   = (int*)  carve((size_t)ET * 4);

  const int TPB = 256;

  // ---- CSR build (dst-sorted incidence, shared by both conv layers) ----
  hipMemsetAsync(cnt, 0, (size_t)NN * 4, stream);
  count_dst<<<(ET + TPB - 1) / TPB, TPB, 0, stream>>>(e_dst, cnt);
  scan_counts<<<1, 1024, 0, stream>>>(cnt, rowst);
  hipMemsetAsync(cnt, 0, (size_t)NN * 4, stream);
  fill_eid<<<(ET + TPB - 1) / TPB, TPB, 0, stream>>>(e_dst, rowst, cnt, eid);

  // ---- conv1: 128 -> 4 heads x 32 ----
  {
    dim3 g(NN / 16, 128 / 32);   // NT=2 -> 32 output cols per wave
    gemm_wmma_f32<128, 128, 2><<<g, 32, 0, stream>>>(x, W1, h1);
  }
  attn_logits<<<(NN * 4 + TPB - 1) / TPB, TPB, 0, stream>>>(h1, as1, ad1, al_s, al_d, NN, 4);
  hipMemsetAsync(denom, 0, (size_t)NN * 4 * 4, stream);
  edge_softmax_num<4><<<((size_t)ET * 4 + TPB - 1) / TPB, TPB, 0, stream>>>(
      e_src, e_dst, al_s, al_d, ex, denom);
  aggregate<4, true><<<NN, 128, 0, stream>>>(e_src, rowst, eid, h1, ex, denom, b1, z1);

  // ---- conv2: 128 -> 8 heads x 32 ----
  {
    dim3 g(NN / 16, 256 / 32);
    gemm_wmma_f32<128, 256, 2><<<g, 32, 0, stream>>>(z1, W2, h2);
  }
  attn_logits<<<(NN * 8 + TPB - 1) / TPB, TPB, 0, stream>>>(h2, as2, ad2, al_s, al_d, NN, 8);
  hipMemsetAsync(denom, 0, (size_t)NN * 8 * 4, stream);
  edge_softmax_num<8><<<((size_t)ET * 8 + TPB - 1) / TPB, TPB, 0, stream>>>(
      e_src, e_dst, al_s, al_d, ex, denom);
  aggregate<8, false><<<NN, 256, 0, stream>>>(e_src, rowst, eid, h2, ex, denom, b2, z2);

  // ---- decode ----
  decode_dot<<<((size_t)2 * EDN * 32 + TPB - 1) / TPB, TPB, 0, stream>>>(z2, pe, ne, out);
}